// CPC_78623671320635
// MI455X (gfx1250) — compile-verified
//
#include <hip/hip_runtime.h>
#include <math.h>

// ---------------- problem constants ----------------
#define BB 128      // batch
#define LL 16384    // sequence length
#define CC 12       // input channels
#define DD 128      // model dim
#define KK 64       // conv kernel
#define STRIDE 64
#define TT 256      // conv output timesteps: (16384-64)/64+1
#define TI 64
#define TO 8
#define SS 183      // TT - TI - (TO+1)
#define RED 768     // C*K reduction for conv
#define LDP 132     // padded LDS row stride for 128-wide tiles
#define LDA 772     // padded LDS row stride for 768-wide conv A tile (772%64==4)

typedef __attribute__((ext_vector_type(2))) float v2f;
typedef __attribute__((ext_vector_type(8))) float v8f;

#if __has_builtin(__builtin_amdgcn_global_load_async_to_lds_b128) && \
    __has_builtin(__builtin_amdgcn_s_wait_asynccnt)
#define USE_ASYNC_LDS 1
// exact parameter types per hipcc diagnostic: non-const v4i (vector_size style),
// global (AS1) source, LDS (AS3) destination
typedef int v4i_ __attribute__((vector_size(4 * sizeof(int))));
typedef __attribute__((address_space(1))) v4i_ gv4i_t;
typedef __attribute__((address_space(3))) v4i_ lv4i_t;
#endif

// D = A(16x4 f32) * B(4x16 f32) + C  — exact f32 matrix op on CDNA5
__device__ __forceinline__ v8f wmma4(v2f a, v2f b, v8f c) {
  return __builtin_amdgcn_wmma_f32_16x16x4_f32(
      /*neg_a=*/false, a, /*neg_b=*/false, b,
      /*c_mod=*/(short)0, c, /*reuse_a=*/false, /*reuse_b=*/false);
}

// A fragment: A[m][k0+k], rows at base + m*ldr (generic ptr: global or LDS)
__device__ __forceinline__ v2f afrag(const float* base, int ldr, int k0, int lane) {
  int m = lane & 15;
  int koff = (lane >> 4) << 1;         // lanes 0-15 -> K {0,1}; 16-31 -> K {2,3}
  const float* p = base + m * ldr + k0 + koff;
  v2f a; a.x = p[0]; a.y = p[1];
  return a;
}

// B fragment for B[k][n] = W[n][k] (i.e. multiply by W^T), rows at base + n*ldw
__device__ __forceinline__ v2f bfragT(const float* base, int ldw, int k0, int lane) {
  int n = lane & 15;
  int koff = (lane >> 4) << 1;
  const float* p = base + n * ldw + k0 + koff;
  v2f b; b.x = p[0]; b.y = p[1];
  return b;
}

// acc += A(16xK) * W(NxK)^T  over K=klen (multiple of 4)
__device__ __forceinline__ v8f gemm_wt(const float* A0, int lda,
                                       const float* W0, int ldw,
                                       int klen, int lane, v8f acc) {
#pragma unroll 4
  for (int k0 = 0; k0 < klen; k0 += 4) {
    v2f a = afrag(A0, lda, k0, lane);
    v2f b = bfragT(W0, ldw, k0, lane);
    acc = wmma4(a, b, acc);
  }
  return acc;
}

__device__ __forceinline__ float sigmoidf_(float x) { return 1.0f / (1.0f + __expf(-x)); }

// ---------------- kernel 1: re-index conv weights ----------------
// Wr[d][k*CC+c] = conv_w[d][c][k]  (so conv reduction matches X's contiguous layout)
__global__ void k_reindex(const float* __restrict__ cw, float* __restrict__ Wr) {
  int i = blockIdx.x * blockDim.x + threadIdx.x;
  if (i >= DD * RED) return;
  int d = i / RED, r = i % RED;
  int k = r / CC, c = r % CC;
  Wr[d * RED + r] = cw[(d * CC + c) * KK + k];
}

// ---------------- kernel 2: conv as WMMA GEMM + bias + relu -> enc[t][b][d] ----
// grid.x = TT * (BB/16) row tiles; 8 waves/block = 8 column tiles (full D).
// A tile (16 x 768 = 48KB of X) is staged into LDS ONCE per block via the
// CDNA5 async global->LDS path, then all 8 waves consume it (8x L2-traffic cut).
__global__ void __launch_bounds__(256) k_conv(const float* __restrict__ X,
                                              const float* __restrict__ Wr,
                                              const float* __restrict__ cb,
                                              float* __restrict__ enc) {
  __shared__ float Atile[16 * LDA];    // 49.4 KB, bank-conflict-free pitch

  int tid = threadIdx.x;
  int lane = tid & 31;
  int wave = tid >> 5;
  int rt = blockIdx.x;                 // row-tile id
  int t = rt >> 3;                     // timestep
  int b0 = (rt & 7) << 4;              // batch base of the 16-row tile
  int d0 = wave << 4;                  // output-channel base

  // A row (t, b): contiguous 768 floats starting at X[b][t*64][0]
  const float* A0 = X + ((size_t)b0 * LL + (size_t)t * STRIDE) * CC;
  const int lda_g = LL * CC;           // stride between consecutive b rows

  // ---- stage A tile: 16 rows x 48 float4-chunks = 3072 chunks, 12 per thread
#pragma unroll
  for (int it = 0; it < 12; ++it) {
    int q = tid + it * 256;            // chunk id 0..3071
    int r = q / 48, c4 = (q % 48) << 2;
    const float* g = A0 + (size_t)r * lda_g + c4;
    float* l = Atile + r * LDA + c4;
#ifdef USE_ASYNC_LDS
    __builtin_amdgcn_global_load_async_to_lds_b128(
        (gv4i_t*)g, (lv4i_t*)l, /*offset=*/0, /*cpol=*/0);
#else
    *(float4*)l = *(const float4*)g;
#endif
  }
#ifdef USE_ASYNC_LDS
  __builtin_amdgcn_s_wait_asynccnt(0);
#endif
  __syncthreads();

  int n = lane & 15;
  float bias = cb[d0 + n];
  v8f acc;
#pragma unroll
  for (int v = 0; v < 8; ++v) acc[v] = bias;

  __builtin_prefetch(Wr + (size_t)d0 * RED, 0, 3);
  acc = gemm_wt(Atile, LDA, Wr + (size_t)d0 * RED, RED, RED, lane, acc);

  int hi = lane >> 4;
#pragma unroll
  for (int v = 0; v < 8; ++v) {
    int m = v + (hi << 3);
    float x = acc[v];
    x = x > 0.0f ? x : 0.0f;           // relu
    enc[((size_t)t * BB + (b0 + m)) * DD + d0 + n] = x;
  }
}

// ---------------- kernel 3: persistent GRU (single workgroup, 32 waves) -------
// LDS: Hs/Rs/Ns/Zs each 128x132 f32 (~264 KB total, dynamic; 320KB WGP budget)
__global__ void __launch_bounds__(1024) k_gru(const float* __restrict__ enc,
                                              const float* __restrict__ Wih,
                                              const float* __restrict__ Whh,
                                              const float* __restrict__ bih,
                                              const float* __restrict__ bhh,
                                              float* __restrict__ ctx,
                                              float* __restrict__ out_hidden) {
  extern __shared__ float smem[];
  float* Hs = smem;                  // h        128 x LDP
  float* Rs = Hs + BB * LDP;         // r gate
  float* Ns = Rs + BB * LDP;         // n gate
  float* Zs = Ns + BB * LDP;         // z gate

  int tid = threadIdx.x;
  int lane = tid & 31;
  int wave = tid >> 5;
  int n = lane & 15;
  int hi = lane >> 4;

  // h = 0
  for (int i = tid; i < BB * LDP; i += 1024) Hs[i] = 0.0f;
  __syncthreads();

  for (int t = 0; t < TT; ++t) {
    const float* Et = enc + (size_t)t * BB * DD;

    // ---- phase 1: r = sigmoid(enc@Wih_r^T + h@Whh_r^T + b)  (cols 0..127)
#pragma unroll
    for (int i = 0; i < 2; ++i) {
      int tile = wave * 2 + i;       // 64 tiles
      int bi = tile >> 3, ji = tile & 7;
      int jb = ji << 4;
      v8f acc;
      float bias = bih[jb + n] + bhh[jb + n];
#pragma unroll
      for (int v = 0; v < 8; ++v) acc[v] = bias;
      __builtin_prefetch(Whh + (size_t)jb * DD, 0, 3);
      acc = gemm_wt(Et + (bi << 4) * DD, DD, Wih + (size_t)jb * DD, DD, DD, lane, acc);
      acc = gemm_wt(Hs + (bi << 4) * LDP, LDP, Whh + (size_t)jb * DD, DD, DD, lane, acc);
#pragma unroll
      for (int v = 0; v < 8; ++v) {
        int m = v + (hi << 3);
        Rs[(size_t)((bi << 4) + m) * LDP + jb + n] = sigmoidf_(acc[v]);
      }
    }
    __syncthreads();

    // ---- phase 2: n = tanh(gi_n + r * gh_n)  (cols 256..383, unfused paths)
#pragma unroll
    for (int i = 0; i < 2; ++i) {
      int tile = wave * 2 + i;
      int bi = tile >> 3, ji = tile & 7;
      int jb = ji << 4;
      int jg = 2 * DD + jb;          // n-gate column base in 3D weights
      v8f ai, ah;
      float bi_n = bih[jg + n], bh_n = bhh[jg + n];
#pragma unroll
      for (int v = 0; v < 8; ++v) { ai[v] = bi_n; ah[v] = bh_n; }
      __builtin_prefetch(Whh + (size_t)jg * DD, 0, 3);
      ai = gemm_wt(Et + (bi << 4) * DD, DD, Wih + (size_t)jg * DD, DD, DD, lane, ai);
      ah = gemm_wt(Hs + (bi << 4) * LDP, LDP, Whh + (size_t)jg * DD, DD, DD, lane, ah);
#pragma unroll
      for (int v = 0; v < 8; ++v) {
        int m = v + (hi << 3);
        size_t idx = (size_t)((bi << 4) + m) * LDP + jb + n;
        float r = Rs[idx];
        Ns[idx] = tanhf(ai[v] + r * ah[v]);
      }
    }

    // ---- phase 3: z = sigmoid(fused sums)  (cols 128..255)
#pragma unroll
    for (int i = 0; i < 2; ++i) {
      int tile = wave * 2 + i;
      int bi = tile >> 3, ji = tile & 7;
      int jb = ji << 4;
      int jg = DD + jb;
      v8f acc;
      float bias = bih[jg + n] + bhh[jg + n];
#pragma unroll
      for (int v = 0; v < 8; ++v) acc[v] = bias;
      __builtin_prefetch(Whh + (size_t)jg * DD, 0, 3);
      acc = gemm_wt(Et + (bi << 4) * DD, DD, Wih + (size_t)jg * DD, DD, DD, lane, acc);
      acc = gemm_wt(Hs + (bi << 4) * LDP, LDP, Whh + (size_t)jg * DD, DD, DD, lane, acc);
#pragma unroll
      for (int v = 0; v < 8; ++v) {
        int m = v + (hi << 3);
        Zs[(size_t)((bi << 4) + m) * LDP + jb + n] = sigmoidf_(acc[v]);
      }
    }
    __syncthreads();

    // ---- update: h = (1-z)*n + z*h ; ctx[t] = h
#pragma unroll
    for (int e = 0; e < 16; ++e) {
      int i = tid * 16 + e;          // 1024*16 = 16384 = 128*128
      int b = i >> 7, d = i & 127;
      size_t idx = (size_t)b * LDP + d;
      float z = Zs[idx], nv = Ns[idx], h = Hs[idx];
      float hn = (1.0f - z) * nv + z * h;
      Hs[idx] = hn;
      ctx[((size_t)t * BB + b) * DD + d] = hn;
    }
    __syncthreads();
  }

  // hidden = hT (1,128,128)
#pragma unroll
  for (int e = 0; e < 16; ++e) {
    int i = tid * 16 + e;
    int b = i >> 7, d = i & 127;
    out_hidden[i] = Hs[(size_t)b * LDP + d];
  }
}

// ---------------- kernel 4: per-(o,s) pred+scores+softmax stats ---------------
// block = 256 (8 waves); LDS: P(128xLDP) + Sc(128xLDP) + lse(128) + red(128)
__global__ void __launch_bounds__(256) k_score(const float* __restrict__ enc,
                                               const float* __restrict__ ctx,
                                               const float* __restrict__ pW,
                                               const float* __restrict__ pb,
                                               float* __restrict__ loss_part,
                                               float* __restrict__ corr_part) {
  extern __shared__ float smem[];
  float* P   = smem;                 // pred tile: P[c][e]
  float* Sc  = P + BB * LDP;         // scores:   Sc[b][c]
  float* lse = Sc + BB * LDP;        // 128
  float* red = lse + BB;             // 128

  int os = blockIdx.x;
  int o = os / SS, s = os % SS;
  int tid = threadIdx.x;
  int lane = tid & 31;
  int wave = tid >> 5;
  int n = lane & 15;
  int hi = lane >> 4;

  const float* Ctx = ctx + ((size_t)(TI + s) * BB) * DD;          // (128 x 128)
  const float* Tgt = enc + ((size_t)(TI + 1 + o + s) * BB) * DD;  // (128 x 128)
  const float* Wo  = pW + (size_t)o * DD * DD;                    // (e x d)

  // ---- step A: P[c][e] = sum_d Ctx[c][d]*Wo[e][d] + pb[o][e]
#pragma unroll
  for (int i = 0; i < 8; ++i) {
    int tile = i * 8 + wave;         // 64 tiles
    int ci = tile >> 3, ei = tile & 7;
    v8f acc;
    float bias = pb[o * DD + (ei << 4) + n];
#pragma unroll
    for (int v = 0; v < 8; ++v) acc[v] = bias;
    __builtin_prefetch(Wo + (size_t)(ei << 4) * DD, 0, 3);
    acc = gemm_wt(Ctx + (ci << 4) * DD, DD, Wo + (size_t)(ei << 4) * DD, DD, DD, lane, acc);
#pragma unroll
    for (int v = 0; v < 8; ++v) {
      int m = v + (hi << 3);
      P[(size_t)((ci << 4) + m) * LDP + (ei << 4) + n] = acc[v];
    }
  }
  __syncthreads();

  // ---- step B: Sc[b][c] = sum_d Tgt[b][d] * P[c][d]   (B operand = P^T from LDS)
#pragma unroll
  for (int i = 0; i < 8; ++i) {
    int tile = i * 8 + wave;
    int bi = tile >> 3, ci = tile & 7;
    v8f acc;
#pragma unroll
    for (int v = 0; v < 8; ++v) acc[v] = 0.0f;
    acc = gemm_wt(Tgt + (bi << 4) * DD, DD, P + (size_t)(ci << 4) * LDP, LDP, DD, lane, acc);
#pragma unroll
    for (int v = 0; v < 8; ++v) {
      int m = v + (hi << 3);
      Sc[(size_t)((bi << 4) + m) * LDP + (ci << 4) + n] = acc[v];
    }
  }
  __syncthreads();

  // ---- step C: per-row logsumexp over c; diag loss contribution
  if (tid < BB) {
    int b = tid;
    float mx = -INFINITY;
    for (int c = 0; c < BB; ++c) mx = fmaxf(mx, Sc[(size_t)b * LDP + c]);
    float sum = 0.0f;
    for (int c = 0; c < BB; ++c) sum += __expf(Sc[(size_t)b * LDP + c] - mx);
    float l = mx + __logf(sum);
    lse[b] = l;
    red[b] = Sc[(size_t)b * LDP + b] - l;   // diag of log_softmax
  }
  __syncthreads();
  if (tid == 0) {
    float acc = 0.0f;
    for (int b = 0; b < BB; ++b) acc += red[b];   // fixed order -> deterministic
    loss_part[os] = -acc;
  }
  __syncthreads();

  // ---- step D: accuracy — per column c, argmax over b of logsm[b][c]
  if (tid < BB) {
    int c = tid;
    float best = -INFINITY; int bi = 0;
    for (int b = 0; b < BB; ++b) {
      float v = Sc[(size_t)b * LDP + c] - lse[b];
      if (v > best) { best = v; bi = b; }
    }
    red[c] = (bi == c) ? 1.0f : 0.0f;
  }
  __syncthreads();
  if (tid == 0) {
    float acc = 0.0f;
    for (int c = 0; c < BB; ++c) acc += red[c];
    corr_part[os] = acc;
  }
}

// ---------------- kernel 5: deterministic final reduction ---------------------
__global__ void k_final(const float* __restrict__ loss_part,
                        const float* __restrict__ corr_part,
                        float* __restrict__ out) {
  if (threadIdx.x == 0 && blockIdx.x == 0) {
    float ls = 0.0f, cs = 0.0f;
    for (int i = 0; i < TO * SS; ++i) { ls += loss_part[i]; cs += corr_part[i]; }
    float denom = (float)(BB * TO * SS);
    out[0] = cs / denom;   // accuracy
    out[1] = ls / denom;   // loss
  }
}

// ---------------- host launcher ----------------
extern "C" void kernel_launch(void* const* d_in, const int* in_sizes, int n_in,
                              void* d_out, int out_size, void* d_ws, size_t ws_size,
                              hipStream_t stream) {
  const float* X      = (const float*)d_in[0];
  const float* conv_w = (const float*)d_in[1];
  const float* conv_b = (const float*)d_in[2];
  const float* W_ih   = (const float*)d_in[3];
  const float* W_hh   = (const float*)d_in[4];
  const float* b_ih   = (const float*)d_in[5];
  const float* b_hh   = (const float*)d_in[6];
  const float* pred_W = (const float*)d_in[7];
  const float* pred_b = (const float*)d_in[8];
  float* out = (float*)d_out;

  float* ws = (float*)d_ws;
  float* Wr        = ws;                              // 128*768
  float* enc       = Wr + (size_t)DD * RED;           // 256*128*128
  float* ctx       = enc + (size_t)TT * BB * DD;      // 256*128*128
  float* loss_part = ctx + (size_t)TT * BB * DD;      // 1464
  float* corr_part = loss_part + TO * SS;             // 1464

  // 1. reindex conv weights
  k_reindex<<<(DD * RED + 255) / 256, 256, 0, stream>>>(conv_w, Wr);

  // 2. conv as GEMM (WMMA f32, async global->LDS staging), relu -> enc
  k_conv<<<TT * (BB / 16), 256, 0, stream>>>(X, Wr, conv_b, enc);

  // 3. GRU (persistent single workgroup, WMMA f32), -> ctx, hidden
  size_t gru_lds = (size_t)4 * BB * LDP * sizeof(float);
  k_gru<<<1, 1024, gru_lds, stream>>>(enc, W_ih, W_hh, b_ih, b_hh, ctx, out + 2);

  // 4. pred + scores + softmax stats per (o,s)
  size_t sc_lds = (size_t)(2 * BB * LDP + 2 * BB) * sizeof(float);
  k_score<<<TO * SS, 256, sc_lds, stream>>>(enc, ctx, pred_W, pred_b,
                                            loss_part, corr_part);

  // 5. deterministic reduction -> accuracy, loss
  k_final<<<1, 64, 0, stream>>>(loss_part, corr_part, out);
}